// TSProjector_65111704207955
// MI455X (gfx1250) — compile-verified
//
#include <hip/hip_runtime.h>
#include <math.h>

#define Bz 32
#define Tz 4096
#define Az 128
#define Sz 100
#define BDz 10
#define Hz 50
#define Dz 30

typedef float v2f __attribute__((ext_vector_type(2)));
typedef float v8f __attribute__((ext_vector_type(8)));

__device__ __forceinline__ v8f wmma_f32_k4(v2f a, v2f b, v8f c) {
  // D = A(16x4,f32) * B(4x16,f32) + C(16x16,f32)
  return __builtin_amdgcn_wmma_f32_16x16x4_f32(false, a, false, b, (short)0, c,
                                               false, false);
}

__device__ __forceinline__ float sigmoidf_(float v) {
  return 1.0f / (1.0f + expf(-v));
}

// ---------------------------------------------------------------------------
// K1: z = mlp3(x) ; allsims = z @ c.  One wave32 per 16-row tile.
//   GEMM1: 16x128 @ 128x16(pad from 10)  -> 32x v_wmma_f32_16x16x4_f32
//   sims : 16x10(pad16) @ 10x112(pad)    -> 7 n-tiles x 3 k-steps WMMA
// Padding is branchless: clamp addresses to valid rows, v_cndmask to zero.
// All GEMM1 fragments are hoisted to registers so the 32 WMMAs issue
// back-to-back instead of load->wait->wmma lockstep.
// ---------------------------------------------------------------------------
__global__ void __launch_bounds__(32) mlp_sims_kernel(
    const float* __restrict__ x, const float* __restrict__ c,
    const float* __restrict__ w1, const float* __restrict__ b1,
    const float* __restrict__ w2, const float* __restrict__ b2,
    const float* __restrict__ w3, const float* __restrict__ b3,
    float* __restrict__ z_out, float* __restrict__ allsims) {
  __shared__ float lz1[16 * 16];
  __shared__ float lz3[16 * 16];
  const int lane  = threadIdx.x;
  const int n     = lane & 15;          // column (N) for B/C/D fragments
  const int khalf = (lane >> 4) * 2;    // K offset for second half-wave
  const int rbase = (lane >> 4) * 8;    // M offset for second half-wave (C/D)
  const long row0 = (long)blockIdx.x * 16;
  const bool ncol = (n < BDz);
  const int nclmp = ncol ? n : 0;       // safe row for padded loads

  // ---- layer 1 via WMMA: hoist all fragments, then 32 chained WMMAs
  const float* xr  = x + (row0 + n) * Az + khalf;   // A: lane holds row M=n
  const float* w1c = w1 + nclmp * Az + khalf;       // B[k][n] = w1[n][k]
  v2f afrag[32], bfrag[32];
#pragma unroll
  for (int q = 0; q < 32; ++q) {
    afrag[q].x = xr[q * 4];
    afrag[q].y = xr[q * 4 + 1];
  }
#pragma unroll
  for (int q = 0; q < 32; ++q) {
    const float bx = w1c[q * 4];
    const float by = w1c[q * 4 + 1];
    bfrag[q].x = ncol ? bx : 0.f;       // v_cndmask, no exec branch
    bfrag[q].y = ncol ? by : 0.f;
  }
  v8f acc = {0.f, 0.f, 0.f, 0.f, 0.f, 0.f, 0.f, 0.f};
#pragma unroll
  for (int q = 0; q < 32; ++q) acc = wmma_f32_k4(afrag[q], bfrag[q], acc);

  float bias1 = b1[nclmp];
  bias1 = ncol ? bias1 : 0.f;
#pragma unroll
  for (int r = 0; r < 8; ++r) acc[r] = fmaxf(acc[r] + bias1, 0.f);
#pragma unroll
  for (int r = 0; r < 8; ++r) lz1[(rbase + r) * 16 + n] = acc[r];
  __syncthreads();

  // ---- layers 2+3 (10x10) per row, lanes 0..15 handle one row each
  if (lane < 16) {
    float zr[BDz], z2v[BDz], z3v[BDz];
#pragma unroll
    for (int k = 0; k < BDz; ++k) zr[k] = lz1[lane * 16 + k];
#pragma unroll
    for (int j = 0; j < BDz; ++j) {
      float s = b2[j];
#pragma unroll
      for (int k = 0; k < BDz; ++k) s += w2[j * BDz + k] * zr[k];
      z2v[j] = fmaxf(s, 0.f);
    }
#pragma unroll
    for (int j = 0; j < BDz; ++j) {
      float s = b3[j];
#pragma unroll
      for (int k = 0; k < BDz; ++k) s += w3[j * BDz + k] * z2v[k];
      z3v[j] = s;
    }
    float* zg = z_out + (row0 + lane) * BDz;
#pragma unroll
    for (int j = 0; j < BDz; ++j) {
      lz3[lane * 16 + j] = z3v[j];
      zg[j] = z3v[j];
    }
#pragma unroll
    for (int j = BDz; j < 16; ++j) lz3[lane * 16 + j] = 0.f;  // K padding
  }
  __syncthreads();

  // ---- sims = z3(16x10) @ c(10x100), 7 N-tiles of 16, K padded to 12
  const int m = lane & 15;
  for (int ns = 0; ns < 7; ++ns) {
    const int sc  = ns * 16 + n;
    const int scc = (sc < Sz) ? sc : 0;          // safe column
    v8f acc2 = {0.f, 0.f, 0.f, 0.f, 0.f, 0.f, 0.f, 0.f};
#pragma unroll
    for (int kk = 0; kk < 12; kk += 4) {
      const int k0  = kk + khalf;
      const int k0a = (k0 < BDz) ? k0 : 0;       // safe rows
      const int k1a = (k0 + 1 < BDz) ? k0 + 1 : 0;
      v2f a;
      a.x = lz3[m * 16 + k0];
      a.y = lz3[m * 16 + k0 + 1];
      const float cx = c[k0a * Sz + scc];
      const float cy = c[k1a * Sz + scc];
      v2f bf;
      bf.x = (k0 < BDz && sc < Sz) ? cx : 0.f;   // v_cndmask
      bf.y = (k0 + 1 < BDz && sc < Sz) ? cy : 0.f;
      acc2 = wmma_f32_k4(a, bf, acc2);
    }
    if (sc < Sz) {
#pragma unroll
      for (int r = 0; r < 8; ++r)
        allsims[(row0 + rbase + r) * Sz + sc] = acc2[r];
    }
  }
}

// ---------------------------------------------------------------------------
// K2: serial DP over t=1..T-1. One wave32 per batch; lane l owns states
// 4l..4l+3 in registers; the s-1 shift is one __shfl_up; backtrack bits are
// packed to dense 128-bit rows via a shfl_xor OR-reduce. Lanes 25..31 are
// given a clamped (valid) load address so the hot loop has NO divergent
// branches; their junk results land in unread bit positions.
// ---------------------------------------------------------------------------
__global__ void __launch_bounds__(32) dp_kernel(
    const float* __restrict__ allsims, unsigned* __restrict__ bt_bits) {
  const int b = blockIdx.x;
  const int l = threadIdx.x;
  const float* base = allsims + (size_t)b * Tz * Sz;
  const int soff = (l < 25) ? 4 * l : 96;        // clamp inactive lanes
  const float* lp = base + soff;

  float4 v0 = *(const float4*)(lp);              // st0 = allsims[:,0]
  float st0 = v0.x, st1 = v0.y, st2 = v0.z, st3 = v0.w;
  const int s0 = 4 * l;
  for (int t = 1; t < Tz; ++t) {
    const float4 sim = *(const float4*)(lp + (size_t)t * Sz);
    if (t + 8 < Tz)  // uniform (scalar) branch
      __builtin_prefetch(lp + (size_t)(t + 8) * Sz, 0, 1);
    float prev = __shfl_up(st3, 1, 32);
    float p0 = (l == 0) ? 0.f : prev;  // reference pads stt2[0] with 0
    unsigned i0 = (p0  > st0) ? 1u : 0u; if (s0     >= t) i0 = 1u;
    unsigned i1 = (st0 > st1) ? 1u : 0u; if (s0 + 1 >= t) i1 = 1u;
    unsigned i2 = (st1 > st2) ? 1u : 0u; if (s0 + 2 >= t) i2 = 1u;
    unsigned i3 = (st2 > st3) ? 1u : 0u; if (s0 + 3 >= t) i3 = 1u;
    float n0 = fmaxf(st0, p0)  + sim.x;
    float n1 = fmaxf(st1, st0) + sim.y;
    float n2 = fmaxf(st2, st1) + sim.z;
    float n3 = fmaxf(st3, st2) + sim.w;
    st0 = n0; st1 = n1; st2 = n2; st3 = n3;
    // pack: bit position of state s inside dword (s>>5) is s&31
    unsigned w = (i0 | (i1 << 1) | (i2 << 2) | (i3 << 3)) << (4 * (l & 7));
    w |= __shfl_xor(w, 1, 8);
    w |= __shfl_xor(w, 2, 8);
    w |= __shfl_xor(w, 4, 8);
    if ((l & 7) == 0) bt_bits[(size_t)(b * Tz + t) * 4 + (l >> 3)] = w;
  }
}

// ---------------------------------------------------------------------------
// K3: backtrace. One wave, lane = batch. Per step: one uint4 load whose
// address depends only on t (pipelinable), then a 4-way register select.
// ---------------------------------------------------------------------------
__global__ void __launch_bounds__(32) backtrack_kernel(
    const unsigned* __restrict__ bt_bits, unsigned char* __restrict__ states) {
  const int b = threadIdx.x;
  int state = Sz - 1;
  unsigned char* sb = states + (size_t)b * Tz;
  sb[Tz - 1] = (unsigned char)state;
  for (int i = Tz - 2; i >= 0; --i) {
    const uint4 w = *(const uint4*)(bt_bits + (size_t)(b * Tz + i + 1) * 4);
    const int d = state >> 5;
    unsigned word = (d == 0) ? w.x : (d == 1) ? w.y : (d == 2) ? w.z : w.w;
    state -= (int)((word >> (state & 31)) & 1u);
    sb[i] = (unsigned char)state;
  }
}

// ---------------------------------------------------------------------------
// K4: segment means. Path is monotone with unit steps from 0 to 99, so each
// state owns one contiguous t-range; thread s sums its range in t-order
// (same accumulation order as the reference scatter-add).
// ---------------------------------------------------------------------------
__global__ void __launch_bounds__(128) segavg_kernel(
    const unsigned char* __restrict__ states, const float* __restrict__ z,
    float* __restrict__ zz) {
  const int b = blockIdx.x;
  const int tid = threadIdx.x;
  __shared__ int start[Sz];
  const unsigned char* st = states + (size_t)b * Tz;
  for (int t = tid; t < Tz; t += 128) {
    int s = st[t];
    if (t == 0 || st[t - 1] != s) start[s] = t;
  }
  __syncthreads();
  if (tid < Sz) {
    const int t0 = start[tid];
    const int t1 = (tid == Sz - 1) ? Tz : start[tid + 1];
    float sum[BDz];
#pragma unroll
    for (int k = 0; k < BDz; ++k) sum[k] = 0.f;
    for (int t = t0; t < t1; ++t) {
      const float* zr = z + ((size_t)b * Tz + t) * BDz;
#pragma unroll
      for (int k = 0; k < BDz; ++k) sum[k] += zr[k];
    }
    const float cnt = (float)(t1 - t0);
    float* o = zz + ((size_t)b * Sz + tid) * BDz;
#pragma unroll
    for (int k = 0; k < BDz; ++k) o[k] = sum[k] / cnt;
  }
}

// ---------------------------------------------------------------------------
// K5: LSTM over 100 steps + MLP head. One workgroup per batch; all weights
// staged in LDS (~54 KB of the 320 KB WGP LDS).
// ---------------------------------------------------------------------------
__global__ void __launch_bounds__(256) lstm_head_kernel(
    const float* __restrict__ zz, const float* __restrict__ w_ih,
    const float* __restrict__ w_hh, const float* __restrict__ b_ih,
    const float* __restrict__ b_hh, const float* __restrict__ wr1,
    const float* __restrict__ br1, const float* __restrict__ wr2,
    const float* __restrict__ br2, const float* __restrict__ wr3,
    const float* __restrict__ br3, float* __restrict__ out) {
  const int b = blockIdx.x;
  const int tid = threadIdx.x;
  __shared__ float s_wih[4 * Hz * BDz];   // 200x10
  __shared__ float s_whh[4 * Hz * Hz];    // 200x50
  __shared__ float s_bsum[4 * Hz];        // 200
  __shared__ float s_x[Sz * BDz];         // 100x10
  __shared__ float s_h[Hz], s_c[Hz], s_g[4 * Hz];
  __shared__ float s_r1[Dz], s_r2[Dz];

  for (int i = tid; i < 4 * Hz * BDz; i += 256) s_wih[i] = w_ih[i];
  for (int i = tid; i < 4 * Hz * Hz; i += 256) s_whh[i] = w_hh[i];
  for (int i = tid; i < 4 * Hz; i += 256) s_bsum[i] = b_ih[i] + b_hh[i];
  for (int i = tid; i < Sz * BDz; i += 256) s_x[i] = zz[(size_t)b * Sz * BDz + i];
  if (tid < Hz) { s_h[tid] = 0.f; s_c[tid] = 0.f; }
  __syncthreads();

  for (int s = 0; s < Sz; ++s) {
    if (tid < 4 * Hz) {
      float g = s_bsum[tid];
      const float* xi = s_x + s * BDz;
      const float* wi = s_wih + tid * BDz;
#pragma unroll
      for (int k = 0; k < BDz; ++k) g += wi[k] * xi[k];
      const float* wh = s_whh + tid * Hz;
#pragma unroll 10
      for (int j = 0; j < Hz; ++j) g += wh[j] * s_h[j];
      s_g[tid] = g;
    }
    __syncthreads();
    if (tid < Hz) {
      const float ig = s_g[tid], fg = s_g[Hz + tid];
      const float gg = s_g[2 * Hz + tid], og = s_g[3 * Hz + tid];
      const float cn = sigmoidf_(fg) * s_c[tid] + sigmoidf_(ig) * tanhf(gg);
      s_c[tid] = cn;
      s_h[tid] = sigmoidf_(og) * tanhf(cn);
    }
    __syncthreads();
  }

  // head: 50 -> 30 -> 30 -> 1
  if (tid < Dz) {
    float r = br1[tid];
#pragma unroll 10
    for (int j = 0; j < Hz; ++j) r += wr1[tid * Hz + j] * s_h[j];
    s_r1[tid] = fmaxf(r, 0.f);
  }
  __syncthreads();
  if (tid < Dz) {
    float r = br2[tid];
#pragma unroll
    for (int j = 0; j < Dz; ++j) r += wr2[tid * Dz + j] * s_r1[j];
    s_r2[tid] = fmaxf(r, 0.f);
  }
  __syncthreads();
  if (tid == 0) {
    float r = br3[0];
#pragma unroll
    for (int j = 0; j < Dz; ++j) r += wr3[j] * s_r2[j];
    out[b] = r;
  }
}

// ---------------------------------------------------------------------------
extern "C" void kernel_launch(void* const* d_in, const int* in_sizes, int n_in,
                              void* d_out, int out_size, void* d_ws,
                              size_t ws_size, hipStream_t stream) {
  const float* x    = (const float*)d_in[0];
  const float* c    = (const float*)d_in[1];
  const float* w1   = (const float*)d_in[2];
  const float* b1   = (const float*)d_in[3];
  const float* w2   = (const float*)d_in[4];
  const float* b2   = (const float*)d_in[5];
  const float* w3   = (const float*)d_in[6];
  const float* b3   = (const float*)d_in[7];
  const float* w_ih = (const float*)d_in[8];
  const float* w_hh = (const float*)d_in[9];
  const float* b_ih = (const float*)d_in[10];
  const float* b_hh = (const float*)d_in[11];
  const float* wr1  = (const float*)d_in[12];
  const float* br1  = (const float*)d_in[13];
  const float* wr2  = (const float*)d_in[14];
  const float* br2  = (const float*)d_in[15];
  const float* wr3  = (const float*)d_in[16];
  const float* br3  = (const float*)d_in[17];
  (void)in_sizes; (void)n_in; (void)out_size; (void)ws_size;
  float* out = (float*)d_out;

  char* ws = (char*)d_ws;
  size_t off = 0;
  auto wsalloc = [&](size_t bytes) -> void* {
    void* p = ws + off;
    off = (off + bytes + 255) & ~(size_t)255;
    return p;
  };
  float* allsims        = (float*)wsalloc((size_t)Bz * Tz * Sz * sizeof(float));   // 52 MB
  float* zbuf           = (float*)wsalloc((size_t)Bz * Tz * BDz * sizeof(float));  // 5.2 MB
  unsigned* bt          = (unsigned*)wsalloc((size_t)Bz * Tz * 4 * sizeof(unsigned)); // 2 MB
  unsigned char* states = (unsigned char*)wsalloc((size_t)Bz * Tz);                // 128 KB
  float* zz             = (float*)wsalloc((size_t)Bz * Sz * BDz * sizeof(float));  // 128 KB

  mlp_sims_kernel<<<dim3((Bz * Tz) / 16), dim3(32), 0, stream>>>(
      x, c, w1, b1, w2, b2, w3, b3, zbuf, allsims);
  dp_kernel<<<dim3(Bz), dim3(32), 0, stream>>>(allsims, bt);
  backtrack_kernel<<<dim3(1), dim3(32), 0, stream>>>(bt, states);
  segavg_kernel<<<dim3(Bz), dim3(128), 0, stream>>>(states, zbuf, zz);
  lstm_head_kernel<<<dim3(Bz), dim3(256), 0, stream>>>(
      zz, w_ih, w_hh, b_ih, b_hh, wr1, br1, wr2, br2, wr3, br3, out);
}